// ContrastiveLoss_28355374088800
// MI455X (gfx1250) — compile-verified
//
#include <hip/hip_runtime.h>
#include <hip/hip_bf16.h>
#include <stdint.h>

#define B_ROWS 65536
#define DDIM   512
#define KANCH  8
#define CEPS   1e-6f
#define CMARGIN 0.8f
#define WAVES_PER_BLOCK 4
#define ROW_BYTES 2048                 // 512 floats
#define NBUF 4                         // quad buffer, prefetch distance 3
#define LDS_PER_WAVE (NBUF * ROW_BYTES)  // 8 KB per wave

typedef float v4f __attribute__((ext_vector_type(4)));

// Issue one 2KB row as 4 async b128 transfers: lane L moves 16B at +L*16,
// chunks at instruction offsets 0/512/1024/1536 (the instruction offset is
// applied to BOTH the global address and the LDS address).
__device__ __forceinline__ void async_row_to_lds(const float* src, int lane,
                                                 uint32_t lds_lane_addr) {
    uint64_t ga = (uint64_t)(uintptr_t)src + (uint64_t)(lane * 16);
    asm volatile(
        "global_load_async_to_lds_b128 %0, %1, off offset:0\n\t"
        "global_load_async_to_lds_b128 %0, %1, off offset:512\n\t"
        "global_load_async_to_lds_b128 %0, %1, off offset:1024\n\t"
        "global_load_async_to_lds_b128 %0, %1, off offset:1536"
        :
        : "v"(lds_lane_addr), "v"(ga)
        : "memory");
}

// Wait until at most n async transfers remain outstanding (in-order counter).
__device__ __forceinline__ void wait_async_le(int n) {
    switch (n) {
        case 0:  asm volatile("s_wait_asynccnt 0"  ::: "memory"); break;
        case 4:  asm volatile("s_wait_asynccnt 4"  ::: "memory"); break;
        case 8:  asm volatile("s_wait_asynccnt 8"  ::: "memory"); break;
        default: asm volatile("s_wait_asynccnt 12" ::: "memory"); break;
    }
}

__device__ __forceinline__ float acc_sqdiff(v4f x, v4f v, float acc) {
    float t;
    t = x.x - v.x + CEPS; acc = fmaf(t, t, acc);
    t = x.y - v.y + CEPS; acc = fmaf(t, t, acc);
    t = x.z - v.z + CEPS; acc = fmaf(t, t, acc);
    t = x.w - v.w + CEPS; acc = fmaf(t, t, acc);
    return acc;
}

__global__ __launch_bounds__(WAVES_PER_BLOCK * 32)
void contrastive_loss_kernel(const float* __restrict__ output1,
                             const float* __restrict__ vectors,
                             const float* __restrict__ feat1,
                             const int*   __restrict__ label,
                             const int*   __restrict__ alpha,
                             float*       __restrict__ loss) {
    extern __shared__ v4f smem4[];
    char* smem = (char*)smem4;

    const int lane = threadIdx.x & 31;
    const int wave = threadIdx.x >> 5;
    const long long b = (long long)blockIdx.x * WAVES_PER_BLOCK + wave;
    if (b >= B_ROWS) return;  // wave-uniform

    // Group-relative LDS byte address of this wave's staging region (generic
    // LDS pointers carry the group offset in the low 32 bits).
    const uint32_t lds_wave = (uint32_t)(uintptr_t)smem + (uint32_t)(wave * LDS_PER_WAVE);
    const uint32_t lds_lane = lds_wave + (uint32_t)(lane * 16);

    // output1 row cached in registers (streamed once -> non-temporal).
    const v4f* xv = (const v4f*)(output1 + (size_t)b * DDIM);
    const v4f x0 = __builtin_nontemporal_load(xv + lane);
    const v4f x1 = __builtin_nontemporal_load(xv + lane + 32);
    const v4f x2 = __builtin_nontemporal_load(xv + lane + 64);
    const v4f x3 = __builtin_nontemporal_load(xv + lane + 96);

    const float* sp[KANCH + 1];
#pragma unroll
    for (int k = 0; k < KANCH; ++k)
        sp[k] = vectors + ((size_t)k * B_ROWS + (size_t)b) * DDIM;
    sp[KANCH] = feat1 + (size_t)b * DDIM;

    // Prime a depth-3 async pipeline: streams 0..2 -> buffers 0..2.
    async_row_to_lds(sp[0], lane, lds_lane + 0 * ROW_BYTES);
    async_row_to_lds(sp[1], lane, lds_lane + 1 * ROW_BYTES);
    async_row_to_lds(sp[2], lane, lds_lane + 2 * ROW_BYTES);

    float sum_anchors = 0.0f;
    float d_feat = 0.0f;

#pragma unroll
    for (int j = 0; j <= KANCH; ++j) {
        if (j + 3 <= KANCH)
            async_row_to_lds(sp[j + 3], lane, lds_lane + (uint32_t)(((j + 3) & (NBUF - 1)) * ROW_BYTES));

        // Streams beyond j still in flight: min(3, KANCH - j); 4 transfers each.
        const int rem = (KANCH - j) < 3 ? (KANCH - j) : 3;
        wait_async_le(rem * 4);

        const v4f* vb = (const v4f*)(smem + wave * LDS_PER_WAVE + (j & (NBUF - 1)) * ROW_BYTES);
        v4f v0 = vb[lane];
        v4f v1 = vb[lane + 32];
        v4f v2 = vb[lane + 64];
        v4f v3 = vb[lane + 96];

        float acc = 0.0f;
        acc = acc_sqdiff(x0, v0, acc);
        acc = acc_sqdiff(x1, v1, acc);
        acc = acc_sqdiff(x2, v2, acc);
        acc = acc_sqdiff(x3, v3, acc);

        // wave32 butterfly reduction
#pragma unroll
        for (int off = 16; off >= 1; off >>= 1)
            acc += __shfl_xor(acc, off, 32);

        float dist = sqrtf(acc);
        if (j < KANCH) sum_anchors += dist; else d_feat = dist;
    }

    const float inv_sqrt_d = 0.04419417382415922f;  // 1/sqrt(512)
    const float alphaF = (float)alpha[0];
    const float ed = (sum_anchors + alphaF * d_feat) * inv_sqrt_d;
    const float marg = ((float)KANCH + alphaF) * CMARGIN;

    if (lane == 0) {
        const float lab = (float)label[b];
        const float h = fmaxf(0.0f, marg - ed);
        const float l = (1.0f - lab) * ed * ed * 0.5f + lab * h * h * 0.5f;
        __builtin_nontemporal_store(l, &loss[b]);
    }
}

extern "C" void kernel_launch(void* const* d_in, const int* in_sizes, int n_in,
                              void* d_out, int out_size, void* d_ws, size_t ws_size,
                              hipStream_t stream) {
    const float* output1 = (const float*)d_in[0];
    const float* vectors = (const float*)d_in[1];
    const float* feat1   = (const float*)d_in[2];
    const int*   label   = (const int*)d_in[3];
    const int*   alpha   = (const int*)d_in[4];
    float* loss = (float*)d_out;

    dim3 grid(B_ROWS / WAVES_PER_BLOCK);
    dim3 block(WAVES_PER_BLOCK * 32);
    size_t shmem = (size_t)WAVES_PER_BLOCK * LDS_PER_WAVE;  // 32 KB / block

    hipLaunchKernelGGL(contrastive_loss_kernel, grid, block, shmem, stream,
                       output1, vectors, feat1, label, alpha, loss);
}